// HyperPatchInvertedResidual_55843164783203
// MI455X (gfx1250) — compile-verified
//
#include <hip/hip_runtime.h>

// ---------------------------------------------------------------------------
// HyperPatchInvertedResidual for MI455X (gfx1250, wave32)
// fp32 path using V_WMMA_F32_16X16X4_F32 for both 1x1 convs.
// Pipeline (BN training-mode stats force global reductions between stages):
//   zero_stats -> stage1(WMMA)+stats -> finalize1 -> stage2(dw3x3)+stats ->
//   finalize2 -> stage3(WMMA)+stats -> finalize3 -> final(residual add)
// ---------------------------------------------------------------------------

typedef __attribute__((ext_vector_type(2))) float v2f;
typedef __attribute__((ext_vector_type(8))) float v8f;

#define B_     8
#define CIN    32
#define HH     256
#define WW     256
#define FH     32
#define FW     32
#define PH     8
#define KH     10
#define HID    64
#define OUTC   32
#define R1_    2048
#define R2_    2624
#define R3_    4672
#define NPATCH (B_ * FH * FW)      // 8192
#define EPS_   1e-5f

// workspace layout (floats)
#define YSZ  ((size_t)NPATCH * HID * 100)   // 52,428,800
#define ZSZ  ((size_t)NPATCH * HID * 64)    // 33,554,432
// O aliases Y (Y is dead after stage2)

__device__ __forceinline__ int reflect_idx(int i, int n) {
    if (i < 0) i = -i;                 // -1 -> 1
    if (i >= n) i = 2 * n - 2 - i;     // n  -> n-2
    return i;
}

__device__ __forceinline__ float relu6f(float v) {
    return fminf(fmaxf(v, 0.0f), 6.0f);
}

// ---------------------------------------------------------------- zero stats
__global__ void zero_stats_kernel(float* stats) {
    int i = threadIdx.x;
    if (i < 640) stats[i] = 0.0f;   // zero whole stats region
}

// ------------------------------------------------------------- BN finalize
__global__ void finalize_bn_kernel(const float* __restrict__ sum,
                                   const float* __restrict__ sq,
                                   float* __restrict__ mean,
                                   float* __restrict__ inv,
                                   int n, float count) {
    int i = threadIdx.x;
    if (i < n) {
        float m = sum[i] / count;
        float v = sq[i] / count - m * m;
        mean[i] = m;
        inv[i]  = rsqrtf(v + EPS_);
    }
}

// --------------------------------------------------- stage 1: 1x1 expand GEMM
// Per patch: Y[64][100] = W1[64][32] * P[32][100]   (P has reflect halo)
__global__ __launch_bounds__(256) void stage1_kernel(
        const float* __restrict__ x, const float* __restrict__ s,
        float* __restrict__ Y, float* __restrict__ gSum, float* __restrict__ gSq) {
    __shared__ float sW1[HID][CIN];    // 64x32
    __shared__ float sP[CIN][112];     // N padded to 7*16, cols>=100 are zero
    __shared__ float sSum[HID], sSq[HID];

    const int patch = blockIdx.x;
    const int b  = patch >> 10;
    const int fy = (patch >> 5) & 31;
    const int fx = patch & 31;
    const int tid = threadIdx.x;

    if (tid < HID) { sSum[tid] = 0.0f; sSq[tid] = 0.0f; }

    for (int i = tid; i < HID * CIN; i += 256) {
        int o = i >> 5, c = i & 31;
        sW1[o][c] = s[(((size_t)b * R3_ + (o * CIN + c)) << 10) + (fy << 5) + fx];
    }
    for (int i = tid; i < CIN * 112; i += 256) {
        int c = i / 112, hw = i % 112;
        float v = 0.0f;
        if (hw < 100) {
            int u = hw / 10, vv = hw % 10;
            int iy = reflect_idx(fy * PH + u - 1, HH);
            int ix = reflect_idx(fx * PH + vv - 1, WW);
            v = x[(((size_t)b * CIN + c) << 16) + (iy << 8) + ix];
        }
        sP[c][hw] = v;
    }
    __syncthreads();

    const int wave  = tid >> 5;
    const int lane  = tid & 31;
    const int lhalf = lane >> 4;   // 0: K={0,1}, 1: K={2,3} (ISA 16x4 f32 A layout)
    const int l15   = lane & 15;

    float* ybase = Y + (size_t)patch * HID * 100;

    for (int t = wave; t < 28; t += 8) {          // 4 Mtiles x 7 Ntiles
        int mt = t / 7, nt = t % 7;
        v8f acc = {};
        for (int k = 0; k < CIN; k += 4) {
            int ka = k + lhalf * 2;
            v2f a, bb;
            a.x  = sW1[mt * 16 + l15][ka];
            a.y  = sW1[mt * 16 + l15][ka + 1];
            bb.x = sP[ka][nt * 16 + l15];
            bb.y = sP[ka + 1][nt * 16 + l15];
            acc = __builtin_amdgcn_wmma_f32_16x16x4_f32(
                    false, a, false, bb, (short)0, acc, false, false);
        }
        int col = nt * 16 + l15;
        bool valid = (col < 100);
#pragma unroll
        for (int r = 0; r < 8; ++r) {
            int m = mt * 16 + lhalf * 8 + r;      // C/D layout: vgpr r -> row r(+8)
            float v = acc[r];
            if (valid) {
                ybase[m * 100 + col] = v;
                atomicAdd(&sSum[m], v);
                atomicAdd(&sSq[m],  v * v);
            }
        }
    }
    __syncthreads();
    if (tid < HID) {
        atomicAdd(&gSum[tid], sSum[tid]);
        atomicAdd(&gSq[tid],  sSq[tid]);
    }
}

// ---------------------------------------------- stage 2: BN1+ReLU6 + dw 3x3
__global__ __launch_bounds__(256) void stage2_kernel(
        const float* __restrict__ s, const float* __restrict__ Y,
        float* __restrict__ Z,
        const float* __restrict__ mean1, const float* __restrict__ inv1,
        const float* __restrict__ g1, const float* __restrict__ b1,
        float* __restrict__ gSum, float* __restrict__ gSq) {
    __shared__ float sY[HID][100];
    __shared__ float sW2[HID][9];
    __shared__ float sSum[HID], sSq[HID];

    const int patch = blockIdx.x;
    const int b  = patch >> 10;
    const int fy = (patch >> 5) & 31;
    const int fx = patch & 31;
    const int tid = threadIdx.x;

    if (tid < HID) { sSum[tid] = 0.0f; sSq[tid] = 0.0f; }

    const float* ybase = Y + (size_t)patch * HID * 100;
    for (int i = tid; i < HID * 100; i += 256) {
        int ch = i / 100, hw = i % 100;
        float v = (ybase[i] - mean1[ch]) * inv1[ch] * g1[ch] + b1[ch];
        sY[ch][hw] = relu6f(v);
    }
    for (int i = tid; i < HID * 9; i += 256) {
        int ch = i / 9, t = i % 9;
        sW2[ch][t] = s[(((size_t)b * R3_ + R1_ + i) << 10) + (fy << 5) + fx];
    }
    __syncthreads();

    float* zbase = Z + (size_t)patch * HID * 64;
    for (int idx = tid; idx < HID * 64; idx += 256) {
        int ch = idx >> 6, px = idx & 63;
        int py = px >> 3, qx = px & 7;
        float acc = 0.0f;
#pragma unroll
        for (int u = 0; u < 3; ++u)
#pragma unroll
            for (int v = 0; v < 3; ++v)
                acc += sW2[ch][u * 3 + v] * sY[ch][(py + u) * 10 + (qx + v)];
        zbase[idx] = acc;
        atomicAdd(&sSum[ch], acc);
        atomicAdd(&sSq[ch],  acc * acc);
    }
    __syncthreads();
    if (tid < HID) {
        atomicAdd(&gSum[tid], sSum[tid]);
        atomicAdd(&gSq[tid],  sSq[tid]);
    }
}

// ------------------------------------------- stage 3: BN2+ReLU6 + project GEMM
// Per patch: O[32][64] = W3[32][64] * Znorm[64][64]
__global__ __launch_bounds__(256) void stage3_kernel(
        const float* __restrict__ s, const float* __restrict__ Z,
        float* __restrict__ O,
        const float* __restrict__ mean2, const float* __restrict__ inv2,
        const float* __restrict__ g2, const float* __restrict__ b2,
        float* __restrict__ gSum, float* __restrict__ gSq) {
    __shared__ float sZ[HID][64];
    __shared__ float sW3[OUTC][HID];
    __shared__ float sSum[OUTC], sSq[OUTC];

    const int patch = blockIdx.x;
    const int b  = patch >> 10;
    const int fy = (patch >> 5) & 31;
    const int fx = patch & 31;
    const int tid = threadIdx.x;

    if (tid < OUTC) { sSum[tid] = 0.0f; sSq[tid] = 0.0f; }

    const float* zbase = Z + (size_t)patch * HID * 64;
    for (int i = tid; i < HID * 64; i += 256) {
        int ch = i >> 6;
        float v = (zbase[i] - mean2[ch]) * inv2[ch] * g2[ch] + b2[ch];
        sZ[ch][i & 63] = relu6f(v);
    }
    for (int i = tid; i < OUTC * HID; i += 256) {
        int o = i >> 6, c = i & 63;
        sW3[o][c] = s[(((size_t)b * R3_ + R2_ + (o * HID + c)) << 10) + (fy << 5) + fx];
    }
    __syncthreads();

    const int wave  = tid >> 5;
    const int lane  = tid & 31;
    const int lhalf = lane >> 4;
    const int l15   = lane & 15;
    const int mt = wave >> 2, nt = wave & 3;      // 2 Mtiles x 4 Ntiles = 8 waves

    v8f acc = {};
    for (int k = 0; k < HID; k += 4) {
        int ka = k + lhalf * 2;
        v2f a, bb;
        a.x  = sW3[mt * 16 + l15][ka];
        a.y  = sW3[mt * 16 + l15][ka + 1];
        bb.x = sZ[ka][nt * 16 + l15];
        bb.y = sZ[ka + 1][nt * 16 + l15];
        acc = __builtin_amdgcn_wmma_f32_16x16x4_f32(
                false, a, false, bb, (short)0, acc, false, false);
    }
    float* obase = O + (size_t)patch * OUTC * 64;
    int col = nt * 16 + l15;
#pragma unroll
    for (int r = 0; r < 8; ++r) {
        int m = mt * 16 + lhalf * 8 + r;
        float v = acc[r];
        obase[m * 64 + col] = v;
        atomicAdd(&sSum[m], v);
        atomicAdd(&sSq[m],  v * v);
    }
    __syncthreads();
    if (tid < OUTC) {
        atomicAdd(&gSum[tid], sSum[tid]);
        atomicAdd(&gSq[tid],  sSq[tid]);
    }
}

// -------------------------------------------------- final: BN3 + residual add
__global__ __launch_bounds__(256) void final_kernel(
        const float* __restrict__ x, const float* __restrict__ O,
        const float* __restrict__ mean3, const float* __restrict__ inv3,
        const float* __restrict__ g3, const float* __restrict__ b3,
        float* __restrict__ out) {
    size_t idx = (size_t)blockIdx.x * 256 + threadIdx.x;   // 2^24 elements
    int xx = (int)(idx & 255);
    int yy = (int)((idx >> 8) & 255);
    int oc = (int)((idx >> 16) & 31);
    int b  = (int)(idx >> 21);
    int fy = yy >> 3, fx = xx >> 3;
    int patch = (b << 10) + (fy << 5) + fx;
    float o = O[(size_t)patch * OUTC * 64 + oc * 64 + ((yy & 7) << 3) + (xx & 7)];
    o = (o - mean3[oc]) * inv3[oc] * g3[oc] + b3[oc];
    out[idx] = x[idx] + o;
}

// ---------------------------------------------------------------------------
extern "C" void kernel_launch(void* const* d_in, const int* in_sizes, int n_in,
                              void* d_out, int out_size, void* d_ws, size_t ws_size,
                              hipStream_t stream) {
    const float* x  = (const float*)d_in[0];
    const float* s  = (const float*)d_in[1];
    const float* g1 = (const float*)d_in[2];
    const float* b1 = (const float*)d_in[3];
    const float* g2 = (const float*)d_in[4];
    const float* b2 = (const float*)d_in[5];
    const float* g3 = (const float*)d_in[6];
    const float* b3 = (const float*)d_in[7];
    float* out = (float*)d_out;

    float* ws = (float*)d_ws;
    float* Y = ws;                 // [NPATCH][64][100]
    float* Z = ws + YSZ;           // [NPATCH][64][64]
    float* O = ws;                 // aliases Y (dead after stage2)
    float* st = ws + YSZ + ZSZ;    // stats region: 640 floats
    float *sum1 = st,       *sq1 = st + 64,  *mean1 = st + 128, *inv1 = st + 192;
    float *sum2 = st + 256, *sq2 = st + 320, *mean2 = st + 384, *inv2 = st + 448;
    float *sum3 = st + 512, *sq3 = st + 544, *mean3 = st + 576, *inv3 = st + 608;

    zero_stats_kernel<<<1, 640, 0, stream>>>(st);

    stage1_kernel<<<NPATCH, 256, 0, stream>>>(x, s, Y, sum1, sq1);
    finalize_bn_kernel<<<1, 64, 0, stream>>>(sum1, sq1, mean1, inv1, HID,
                                             (float)((size_t)NPATCH * 100));
    stage2_kernel<<<NPATCH, 256, 0, stream>>>(s, Y, Z, mean1, inv1, g1, b1,
                                              sum2, sq2);
    finalize_bn_kernel<<<1, 64, 0, stream>>>(sum2, sq2, mean2, inv2, HID,
                                             (float)((size_t)NPATCH * 64));
    stage3_kernel<<<NPATCH, 256, 0, stream>>>(s, Z, O, mean2, inv2, g2, b2,
                                              sum3, sq3);
    finalize_bn_kernel<<<1, 64, 0, stream>>>(sum3, sq3, mean3, inv3, OUTC,
                                             (float)((size_t)NPATCH * 64));
    final_kernel<<<(1 << 24) / 256, 256, 0, stream>>>(x, O, mean3, inv3, g3, b3,
                                                      out);
    (void)in_sizes; (void)n_in; (void)out_size; (void)ws_size;
}